// EGNN_661424963983
// MI455X (gfx1250) — compile-verified
//
#include <hip/hip_runtime.h>

// EGNN fused kernel for MI455X (gfx1250, wave32, WMMA).
// One workgroup (256 threads = 8 wave32) per batch graph; all GEMMs run on
// V_WMMA_F32_16X16X4_F32 (exact fp32 matrix math, matches fp32 reference).

#define B_   128
#define N_   64
#define D_   3
#define H_   128
#define TE_  64
#define L_   4

// LDS strides chosen for bank-conflict freedom on a 64-bank LDS:
//  - A-side buffers (row-major [64][128]) use stride 132 (132 % 64 == 4):
//    the 16x4 A-fragment b64 loads hit all 64 banks exactly once per wave.
//  - B-side weight buffer uses stride 136 (136 % 64 == 8): the two lane
//    halves (K and K+2 rows) land in disjoint 16-bank blocks.
#define LDA 132
#define LDB 136

typedef float v2f __attribute__((ext_vector_type(2)));
typedef float v8f __attribute__((ext_vector_type(8)));

struct EgnnParams {
  const float* t;
  const float* x;
  const float* embed_w;
  const float* embed_b;
  const float* ew1[L_]; const float* eb1[L_];
  const float* ew2[L_]; const float* eb2[L_];
  const float* cw1[L_]; const float* cb1[L_];
  const float* cw2[L_];
  const float* nw1[L_]; const float* nb1[L_];
  const float* nw2[L_]; const float* nb2[L_];
  float* out;
  float* msgws;   // [B, N, H] message scratch in global ws
};

__device__ __forceinline__ v8f vzero8() {
  v8f z = {0.f, 0.f, 0.f, 0.f, 0.f, 0.f, 0.f, 0.f};
  return z;
}

__device__ __forceinline__ float silu_f(float v) {
  return v / (1.0f + __expf(-v));
}

__device__ __forceinline__ v8f wmma4(v2f a, v2f b, v8f c) {
#if defined(__HIP_DEVICE_COMPILE__)
#if __has_builtin(__builtin_amdgcn_wmma_f32_16x16x4_f32)
  // 8-arg pattern: (neg_a, A, neg_b, B, c_mod, C, reuse_a, reuse_b)
  return __builtin_amdgcn_wmma_f32_16x16x4_f32(false, a, false, b, (short)0, c,
                                               false, false);
#else
#error "device pass: __builtin_amdgcn_wmma_f32_16x16x4_f32 not available"
#endif
#else
  // Host compilation pass parses but never codegens device functions.
  (void)a; (void)b;
  return c;
#endif
}

// acc[t] += A[16*mt .. +16)[0..K) * B[0..K)[64*nh + 16*t .. +16)
// A layout (16x4 f32 frag): lane l holds A[m = l&15][k = kk + 2*(l>>4) + {0,1}]
// B layout (4x16 f32 frag): lane l, vgpr v holds B[kk + 2*(l>>4) + v][n = l&15]
__device__ __forceinline__ void gemm_acc(const float* __restrict__ Asm,
                                         const float* __restrict__ Bsm,
                                         int K, v8f acc[4],
                                         int lane, int mt, int nh)
{
  const int mrow = lane & 15;
  const int khi  = (lane >> 4) << 1;          // 0 or 2
  const float* ap  = Asm + (mt * 16 + mrow) * LDA + khi;
  const float* bp0 = Bsm + khi * LDB + nh * 64 + mrow;
#pragma unroll 2
  for (int kk = 0; kk < K; kk += 4) {
    v2f a = *(const v2f*)(ap + kk);
    const float* bp = bp0 + kk * LDB;
#pragma unroll
    for (int t = 0; t < 4; ++t) {
      v2f b;
      b.x = bp[t * 16];
      b.y = bp[t * 16 + LDB];
      acc[t] = wmma4(a, b, acc[t]);
    }
  }
}

// C/D layout: lane l, vgpr r -> C[m = 16*mt + r + 8*(l>>4)][n = 64*nh + 16*t + (l&15)]
template <int ACT /*silu*/, int RES /*add into dest*/>
__device__ __forceinline__ void epilogue(float* __restrict__ Csm, const v8f acc[4],
                                         int lane, int mt, int nh,
                                         const float* __restrict__ bias, int maskRow)
{
  const int c0    = nh * 64 + (lane & 15);
  const int mbase = mt * 16 + (lane >> 4) * 8;
#pragma unroll
  for (int t = 0; t < 4; ++t) {
    const int c   = c0 + t * 16;
    const float bb = bias ? bias[c] : 0.0f;
#pragma unroll
    for (int r = 0; r < 8; ++r) {
      const int m = mbase + r;
      float v = acc[t][r] + bb;
      if (ACT) v = silu_f(v);
      if (maskRow >= 0 && m == maskRow) v = 0.0f;   // kill self-edge row
      if (RES) Csm[m * LDA + c] += v;
      else     Csm[m * LDA + c]  = v;
    }
  }
}

__global__ __launch_bounds__(256, 1)
void egnn_fused(EgnnParams P)
{
  // ---- LDS arena: 139264 + 5*33792 + ~7.4K = 315,648 B (< 320 KB/WGP) ----
  __shared__ __align__(16) float Wb[256 * LDB];   // weights: [256][128] padded
  __shared__ __align__(16) float Hs[N_ * LDA];    // node features h
  __shared__ __align__(16) float As[N_ * LDA];    // h@Wa   / node hidden
  __shared__ __align__(16) float Bs[N_ * LDA];    // h@Wb
  __shared__ __align__(16) float Xs[N_ * LDA];    // edge act x1 / u
  __shared__ __align__(16) float Ms[N_ * LDA];    // edge m   / msg reload
  __shared__ float wr_s[H_], eb1_s[H_], eb2_s[H_], cb1_s[H_], cw2_s[H_];
  __shared__ float nb1_s[H_], nb2_s[H_];
  __shared__ float temb[TE_];
  __shared__ float pos_s[N_ * 4], pdel_s[N_ * 4], rel_s[N_ * 4];
  __shared__ float dsq_s[N_], cw_s[N_];

  const int b    = blockIdx.x;
  const int tid  = threadIdx.x;
  const int lane = tid & 31;
  const int wv   = tid >> 5;    // 0..7
  const int mt   = wv & 3;      // M tile (16 rows)
  const int nh   = wv >> 2;     // N half (64 cols)

  // ---- prologue: sinusoidal time embedding -> h0, broadcast to all nodes ----
  {
    const float tval = P.t[b];
    if (tid < TE_ / 2) {
      const float f = __expf(-9.210340371976184f * ((float)tid / (float)(TE_ / 2)));
      const float a = tval * f;
      temb[tid]           = __sinf(a);
      temb[tid + TE_ / 2] = __cosf(a);
    }
    if (tid < N_) {
      const float* xp = P.x + (size_t)b * (N_ * D_) + tid * D_;
      pos_s[tid * 4 + 0] = xp[0];
      pos_s[tid * 4 + 1] = xp[1];
      pos_s[tid * 4 + 2] = xp[2];
      pos_s[tid * 4 + 3] = 0.f;
    }
    __syncthreads();
    if (tid < H_) {
      float acc = P.embed_b[tid];
      for (int k = 0; k < TE_; ++k) acc += temb[k] * P.embed_w[k * H_ + tid];
      for (int i = 0; i < N_; ++i) Hs[i * LDA + tid] = acc;
    }
    __syncthreads();
  }

  for (int l = 0; l < L_; ++l) {
    // ---- stage ew1: rows 0..127 = W_a (src half), 128..255 = W_b (dst half),
    //      row 256 = dist^2 weight; plus all per-layer biases ----
    {
      const float* w = P.ew1[l];
      for (int idx = tid; idx < 256 * H_; idx += 256)
        Wb[(idx >> 7) * LDB + (idx & 127)] = w[idx];
      if (tid < H_) {
        wr_s[tid]  = w[256 * H_ + tid];
        eb1_s[tid] = P.eb1[l][tid];
        eb2_s[tid] = P.eb2[l][tid];
        cb1_s[tid] = P.cb1[l][tid];
        cw2_s[tid] = P.cw2[l][tid];
        nb1_s[tid] = P.nb1[l][tid];
        nb2_s[tid] = P.nb2[l][tid];
      }
    }
    __syncthreads();

    // ---- factorized first edge GEMM: A = h@W_a, Bv = h@W_b  (64x128x128) ----
    {
      v8f acc[4] = {vzero8(), vzero8(), vzero8(), vzero8()};
      gemm_acc(Hs, Wb, H_, acc, lane, mt, nh);
      epilogue<0, 0>(As, acc, lane, mt, nh, nullptr, -1);
    }
    {
      v8f acc[4] = {vzero8(), vzero8(), vzero8(), vzero8()};
      gemm_acc(Hs, Wb + 128 * LDB, H_, acc, lane, mt, nh);
      epilogue<0, 0>(Bs, acc, lane, mt, nh, nullptr, -1);
    }
    __syncthreads();

    // ---- restage: ew2 -> rows 0..127, cw1 -> rows 128..255 ----
    {
      const float* w2 = P.ew2[l];
      const float* c1 = P.cw1[l];
      for (int idx = tid; idx < H_ * H_; idx += 256) {
        const int k = idx >> 7, c = idx & 127;
        Wb[k * LDB + c]         = w2[idx];
        Wb[(k + 128) * LDB + c] = c1[idx];
      }
    }
    __syncthreads();

    // ---- edge phase: loop over destination node j (64 src rows at a time) ----
    for (int j = 0; j < N_; ++j) {
      if (tid < N_) {
        const float rx = pos_s[tid * 4 + 0] - pos_s[j * 4 + 0];
        const float ry = pos_s[tid * 4 + 1] - pos_s[j * 4 + 1];
        const float rz = pos_s[tid * 4 + 2] - pos_s[j * 4 + 2];
        rel_s[tid * 4 + 0] = rx;
        rel_s[tid * 4 + 1] = ry;
        rel_s[tid * 4 + 2] = rz;
        dsq_s[tid] = rx * rx + ry * ry + rz * rz;
      }
      __syncthreads();
      // x1 = silu(A_i + Bv_j + dsq_i * wr + eb1)
      for (int e = tid; e < N_ * H_; e += 256) {
        const int i = e >> 7, k = e & 127;
        const float v = As[i * LDA + k] + Bs[j * LDA + k] + dsq_s[i] * wr_s[k] + eb1_s[k];
        Xs[i * LDA + k] = silu_f(v);
      }
      __syncthreads();
      // m = silu(x1 @ ew2 + eb2); zero self-edge row j
      {
        v8f acc[4] = {vzero8(), vzero8(), vzero8(), vzero8()};
        gemm_acc(Xs, Wb, H_, acc, lane, mt, nh);
        epilogue<1, 0>(Ms, acc, lane, mt, nh, eb2_s, j);
      }
      __syncthreads();
      // u = silu(m @ cw1 + cb1) -> Xs (reuse)
      {
        v8f acc[4] = {vzero8(), vzero8(), vzero8(), vzero8()};
        gemm_acc(Ms, Wb + 128 * LDB, H_, acc, lane, mt, nh);
        epilogue<1, 0>(Xs, acc, lane, mt, nh, cb1_s, -1);
      }
      __syncthreads();
      // cw_i = u_i . cw2 (threads 0..63, skewed k to dodge bank conflicts)
      // msg_j = column-sum of m (threads 128..255) -> global ws
      if (tid < N_) {
        float s = 0.f;
        for (int k0 = 0; k0 < H_; ++k0) {
          const int k = (k0 + 2 * tid) & 127;
          s += Xs[tid * LDA + k] * cw2_s[k];
        }
        cw_s[tid] = s;
      } else if (tid >= 128) {
        const int c = tid - 128;
        float s = 0.f;
#pragma unroll 4
        for (int i = 0; i < N_; ++i) s += Ms[i * LDA + c];
        P.msgws[((size_t)b * N_ + j) * H_ + c] = s;
      }
      __syncthreads();
      // pos delta for dst j: sum_i cw_i * rel_i  (rel_jj == 0)
      if (tid < D_) {
        float s = 0.f;
        for (int i = 0; i < N_; ++i) s += cw_s[i] * rel_s[i * 4 + tid];
        pdel_s[j * 4 + tid] = s;
      }
      __syncthreads();
    }

    // ---- node phase: stage nw1 (256x128) + reload msg ----
    {
      const float* w = P.nw1[l];
      for (int idx = tid; idx < 256 * H_; idx += 256)
        Wb[(idx >> 7) * LDB + (idx & 127)] = w[idx];
      for (int idx = tid; idx < N_ * H_; idx += 256) {
        const int i = idx >> 7, c = idx & 127;
        Ms[i * LDA + c] = P.msgws[((size_t)b * N_ + i) * H_ + c];
      }
    }
    __syncthreads();
    // hidden = silu([h | msg] @ nw1 + nb1) -> As  (K = 256, two K-halves)
    {
      v8f acc[4] = {vzero8(), vzero8(), vzero8(), vzero8()};
      gemm_acc(Hs, Wb, H_, acc, lane, mt, nh);
      gemm_acc(Ms, Wb + 128 * LDB, H_, acc, lane, mt, nh);
      epilogue<1, 0>(As, acc, lane, mt, nh, nb1_s, -1);
    }
    __syncthreads();
    // stage nw2 over nw1's space
    {
      const float* w = P.nw2[l];
      for (int idx = tid; idx < H_ * H_; idx += 256)
        Wb[(idx >> 7) * LDB + (idx & 127)] = w[idx];
    }
    __syncthreads();
    // h += hidden @ nw2 + nb2  (residual)
    {
      v8f acc[4] = {vzero8(), vzero8(), vzero8(), vzero8()};
      gemm_acc(As, Wb, H_, acc, lane, mt, nh);
      epilogue<0, 1>(Hs, acc, lane, mt, nh, nb2_s, -1);
    }
    // apply this layer's coordinate update
    if (tid < N_) {
      pos_s[tid * 4 + 0] += pdel_s[tid * 4 + 0];
      pos_s[tid * 4 + 1] += pdel_s[tid * 4 + 1];
      pos_s[tid * 4 + 2] += pdel_s[tid * 4 + 2];
    }
    __syncthreads();
  }

  // ---- epilogue: dx = (pos - x0) minus per-batch center of mass ----
  if (tid < D_) {
    float s = 0.f;
    for (int i = 0; i < N_; ++i)
      s += pos_s[i * 4 + tid] - P.x[(size_t)b * (N_ * D_) + i * D_ + tid];
    dsq_s[tid] = s * (1.0f / (float)N_);
  }
  __syncthreads();
  if (tid < N_) {
    const float* xp = P.x + (size_t)b * (N_ * D_) + tid * D_;
    float* op = P.out + (size_t)b * (N_ * D_) + tid * D_;
    op[0] = pos_s[tid * 4 + 0] - xp[0] - dsq_s[0];
    op[1] = pos_s[tid * 4 + 1] - xp[1] - dsq_s[1];
    op[2] = pos_s[tid * 4 + 2] - xp[2] - dsq_s[2];
  }
}

extern "C" void kernel_launch(void* const* d_in, const int* in_sizes, int n_in,
                              void* d_out, int out_size, void* d_ws, size_t ws_size,
                              hipStream_t stream)
{
  (void)out_size; (void)ws_size;
  EgnnParams P;
  auto F = [&](int i) { return (const float*)d_in[i]; };

  // Detect input flattening from sizes:
  //  A) insertion-order recursive walk: t, x, embed_w, embed_b, [ew1,eb1,ew2,eb2,cw1,cb1,cw2,nw1,nb1,nw2,nb2]x4
  //  B) jax pytree (sorted keys):       embed_b, embed_w, [cb1,cw1,cw2,eb1,eb2,ew1,ew2,nb1,nb2,nw1,nw2]x4, t, x
  const bool insertion = (n_in > 2) && (in_sizes[1] == B_ * N_ * D_);
  if (insertion) {
    P.t = F(0); P.x = F(1); P.embed_w = F(2); P.embed_b = F(3);
    for (int l = 0; l < L_; ++l) {
      const int base = 4 + 11 * l;
      P.ew1[l] = F(base + 0);  P.eb1[l] = F(base + 1);
      P.ew2[l] = F(base + 2);  P.eb2[l] = F(base + 3);
      P.cw1[l] = F(base + 4);  P.cb1[l] = F(base + 5);
      P.cw2[l] = F(base + 6);
      P.nw1[l] = F(base + 7);  P.nb1[l] = F(base + 8);
      P.nw2[l] = F(base + 9);  P.nb2[l] = F(base + 10);
    }
  } else {
    P.embed_b = F(0); P.embed_w = F(1);
    for (int l = 0; l < L_; ++l) {
      const int base = 2 + 11 * l;
      P.cb1[l] = F(base + 0); P.cw1[l] = F(base + 1); P.cw2[l] = F(base + 2);
      P.eb1[l] = F(base + 3); P.eb2[l] = F(base + 4);
      P.ew1[l] = F(base + 5); P.ew2[l] = F(base + 6);
      P.nb1[l] = F(base + 7); P.nb2[l] = F(base + 8);
      P.nw1[l] = F(base + 9); P.nw2[l] = F(base + 10);
    }
    P.t = F(2 + 11 * L_);
    P.x = F(3 + 11 * L_);
  }
  P.out   = (float*)d_out;
  P.msgws = (float*)d_ws;   // needs B*N*H*4 = 4 MB

  egnn_fused<<<dim3(B_), dim3(256), 0, stream>>>(P);
}